// DeformableConvLayer_89421219103701
// MI455X (gfx1250) — compile-verified
//
#include <hip/hip_runtime.h>
#include <hip/hip_bf16.h>

typedef __attribute__((ext_vector_type(2))) float v2f;
typedef __attribute__((ext_vector_type(8))) float v8f;

#define BB   8
#define CIN  64
#define HH   128
#define WW   128
#define HWSZ (HH * WW)
#define COUT 64
#define KK   9
#define TN   64     // pixels per block (half a row)
#define KSTAGE 16

// ---------------------------------------------------------------------------
// Kernel 0: repack w[oc][ci][kk] -> wpack[kk][ci][oc] so WMMA A-fragment loads
// are contiguous in oc across lanes 0..15.
// ---------------------------------------------------------------------------
__global__ void repack_w_kernel(const float* __restrict__ w, float* __restrict__ wpack) {
    int i = blockIdx.x * 256 + threadIdx.x;
    if (i >= KK * CIN * COUT) return;
    int oc = i & 63;
    int ci = (i >> 6) & 63;
    int kk = i >> 12;
    wpack[i] = w[(oc * CIN + ci) * KK + kk];
}

// ---------------------------------------------------------------------------
// Kernel 1: 18-channel 3x3 offset conv (stride1/pad1), b_off added.
// One block per (b,h), 128 threads = w. w_off staged in LDS as [ci*9+tap][c].
// ---------------------------------------------------------------------------
__global__ __launch_bounds__(128) void offset_conv_kernel(
        const float* __restrict__ x, const float* __restrict__ w_off,
        const float* __restrict__ b_off, float* __restrict__ off_ws) {
    __shared__ float sw[CIN * KK * 18];   // 41.5 KB
    const int bh = blockIdx.x;
    const int b  = bh >> 7;
    const int h  = bh & (HH - 1);
    const int wq = threadIdx.x;

    for (int i = threadIdx.x; i < CIN * KK * 18; i += 128) {
        int c = i % 18;
        int rest = i / 18;
        int tap = rest % KK;
        int ci  = rest / KK;
        sw[i] = w_off[(c * CIN + ci) * KK + tap];
    }
    __syncthreads();

    float acc[18];
#pragma unroll
    for (int c = 0; c < 18; ++c) acc[c] = b_off[c];

    const float* xb = x + (size_t)b * CIN * HWSZ;
    for (int ci = 0; ci < CIN; ++ci) {
        const float* xc = xb + ci * HWSZ;
#pragma unroll
        for (int ky = 0; ky < 3; ++ky) {
            int y = h + ky - 1;
            bool yv = (y >= 0) && (y < HH);
#pragma unroll
            for (int kx = 0; kx < 3; ++kx) {
                int xw = wq + kx - 1;
                float v = (yv && xw >= 0 && xw < WW) ? xc[y * WW + xw] : 0.0f;
                const float* ww = &sw[(ci * KK + ky * 3 + kx) * 18];
#pragma unroll
                for (int c = 0; c < 18; ++c) acc[c] = fmaf(v, ww[c], acc[c]);
            }
        }
    }
#pragma unroll
    for (int c = 0; c < 18; ++c)
        off_ws[(((size_t)b * 18 + c) * HH + h) * WW + wq] = acc[c];
}

// ---------------------------------------------------------------------------
// Kernel 2: fused bilinear-gather + implicit-im2col GEMM via f32 WMMA.
// Block: 256 thr = 8 waves; tile = 64 Cout x 64 pixels (one (b,h) row half).
// Wave (mtile = wave&3, npair = wave>>2) owns a 16x32 slab -> 2 accumulators.
// K loop: 9 taps x 64 channels; cols staged 16-K at a time in LDS.
// ---------------------------------------------------------------------------
__global__ __launch_bounds__(256) void deform_gemm_kernel(
        const float* __restrict__ x, const float* __restrict__ off_ws,
        const float* __restrict__ wpack, float* __restrict__ out) {
    __shared__ float s_wgt[4][TN];
    __shared__ int   s_adr[4][TN];
    __shared__ float s_cols[KSTAGE][TN];

    const int tile = blockIdx.x;
    const int pix0 = tile * TN;
    const int b  = pix0 >> 14;          // / (H*W)
    const int h  = (pix0 >> 7) & (HH - 1);
    const int w0 = pix0 & (WW - 1);     // 0 or 64

    const int tid   = threadIdx.x;
    const int lane  = tid & 31;
    const int wave  = tid >> 5;
    const int mtile = wave & 3;         // Cout/16 tile
    const int npair = wave >> 2;        // pixel 32-column pair
    const int mlane = lane & 15;
    const int khalf = lane >> 4;        // A/B VGPR lane-half -> K slot {0,2}/{1,3}
    const int pix   = tid & 63;
    const int cil   = tid >> 6;         // 0..3: K-lane group for gathering

    v8f acc0 = {};
    v8f acc1 = {};

    const float* xbase = x + (size_t)b * CIN * HWSZ;

    for (int kk = 0; kk < KK; ++kk) {
        __syncthreads();                // protect s_wgt/s_adr from previous tap
        if (tid < TN) {
            int wq = w0 + tid;
            float dy = off_ws[(((size_t)b * 18 + 2 * kk) * HH + h) * WW + wq];
            float dx = off_ws[(((size_t)b * 18 + 2 * kk + 1) * HH + h) * WW + wq];
            float py = (float)(h - 1 + kk / 3) + dy;
            float px = (float)(wq - 1 + kk % 3) + dx;
            float y0f = floorf(py), x0f = floorf(px);
            int y0 = (int)y0f, x0 = (int)x0f;
            int y1 = y0 + 1,  x1 = x0 + 1;
            float wy1 = py - y0f, wy0 = 1.0f - wy1;
            float wx1 = px - x0f, wx0 = 1.0f - wx1;
            bool y0v = (y0 >= 0) && (y0 < HH);
            bool y1v = (y1 >= 0) && (y1 < HH);
            bool x0v = (x0 >= 0) && (x0 < WW);
            bool x1v = (x1 >= 0) && (x1 < WW);
            int y0c = y0 < 0 ? 0 : (y0 > HH - 1 ? HH - 1 : y0);
            int y1c = y1 < 0 ? 0 : (y1 > HH - 1 ? HH - 1 : y1);
            int x0c = x0 < 0 ? 0 : (x0 > WW - 1 ? WW - 1 : x0);
            int x1c = x1 < 0 ? 0 : (x1 > WW - 1 ? WW - 1 : x1);
            s_wgt[0][tid] = wy0 * wx0 * ((y0v && x0v) ? 1.0f : 0.0f);
            s_wgt[1][tid] = wy0 * wx1 * ((y0v && x1v) ? 1.0f : 0.0f);
            s_wgt[2][tid] = wy1 * wx0 * ((y1v && x0v) ? 1.0f : 0.0f);
            s_wgt[3][tid] = wy1 * wx1 * ((y1v && x1v) ? 1.0f : 0.0f);
            s_adr[0][tid] = y0c * WW + x0c;
            s_adr[1][tid] = y0c * WW + x1c;
            s_adr[2][tid] = y1c * WW + x0c;
            s_adr[3][tid] = y1c * WW + x1c;
        }
        __syncthreads();

        // per-thread gather metadata for its pixel (reused across all 64 ci)
        const float g0 = s_wgt[0][pix], g1 = s_wgt[1][pix];
        const float g2 = s_wgt[2][pix], g3 = s_wgt[3][pix];
        const int   a0 = s_adr[0][pix], a1 = s_adr[1][pix];
        const int   a2 = s_adr[2][pix], a3 = s_adr[3][pix];

        for (int ci0 = 0; ci0 < CIN; ci0 += KSTAGE) {
            // Stage KSTAGE x 64 cols values; each thread produces 4.
#pragma unroll
            for (int j = 0; j < 4; ++j) {
                int kl = cil + 4 * j;
                const float* xp = xbase + (size_t)(ci0 + kl) * HWSZ;
                float v = g0 * xp[a0] + g1 * xp[a1] + g2 * xp[a2] + g3 * xp[a3];
                s_cols[kl][pix] = v;
            }
            __syncthreads();

#pragma unroll
            for (int ks = 0; ks < KSTAGE; ks += 4) {
                int ci = ci0 + ks;
                // A fragment (16x4 f32): v0 <- K{0,2}, v1 <- K{1,3} by lane half
                const float* wp = wpack + ((size_t)kk * CIN + ci) * COUT
                                        + mtile * 16 + mlane;
                v2f afrag;
                afrag.x = wp[(2 * khalf)     * COUT];
                afrag.y = wp[(2 * khalf + 1) * COUT];
                // B fragments (4x16 f32), mirrored K striping
                int n0 = npair * 32 + mlane;
                v2f bf0, bf1;
                bf0.x = s_cols[ks + 2 * khalf][n0];
                bf0.y = s_cols[ks + 2 * khalf + 1][n0];
                bf1.x = s_cols[ks + 2 * khalf][n0 + 16];
                bf1.y = s_cols[ks + 2 * khalf + 1][n0 + 16];

                acc0 = __builtin_amdgcn_wmma_f32_16x16x4_f32(
                    false, afrag, false, bf0, (short)0, acc0, false, false);
                acc1 = __builtin_amdgcn_wmma_f32_16x16x4_f32(
                    false, afrag, false, bf1, (short)0, acc1, false, false);
            }
            __syncthreads();            // s_cols reuse
        }
    }

    // Store D: VGPR r -> M = r (lanes 0..15) / r+8 (lanes 16..31), N = lane&15
    const int ncol = w0 + npair * 32 + mlane;
#pragma unroll
    for (int r = 0; r < 8; ++r) {
        int oc = mtile * 16 + r + 8 * khalf;
        size_t base = (((size_t)b * COUT + oc) * HH + h) * WW;
        out[base + ncol]      = acc0[r];
        out[base + ncol + 16] = acc1[r];
    }
}

extern "C" void kernel_launch(void* const* d_in, const int* in_sizes, int n_in,
                              void* d_out, int out_size, void* d_ws, size_t ws_size,
                              hipStream_t stream) {
    (void)in_sizes; (void)n_in; (void)out_size; (void)ws_size;
    const float* x     = (const float*)d_in[0];
    const float* w_off = (const float*)d_in[1];
    const float* b_off = (const float*)d_in[2];
    const float* w     = (const float*)d_in[3];
    float* out = (float*)d_out;

    float* off_ws = (float*)d_ws;                             // 8*18*128*128 floats = 9.4 MB
    float* wpack  = off_ws + (size_t)BB * 18 * HH * WW;       // + 147 KB

    repack_w_kernel<<<(KK * CIN * COUT + 255) / 256, 256, 0, stream>>>(w, wpack);
    offset_conv_kernel<<<BB * HH, 128, 0, stream>>>(x, w_off, b_off, off_ws);
    deform_gemm_kernel<<<(BB * HWSZ) / TN, 256, 0, stream>>>(x, off_ws, wpack, out);
}